// xLSTM_46282567582044
// MI455X (gfx1250) — compile-verified
//
#include <hip/hip_runtime.h>
#include <hip/hip_bf16.h>

// ---------------------------------------------------------------------------
// Sizes (fixed by the reference)
// ---------------------------------------------------------------------------
#define B_   256
#define S_   512
#define F_   256
#define L_   20
#define T_   257          // F_+1
#define NSEQ 512          // 2*B_
#define G_   80           // 4*L_

typedef float v2f __attribute__((ext_vector_type(2)));
typedef float v8f __attribute__((ext_vector_type(8)));
typedef int   v4i __attribute__((ext_vector_type(4)));

#define AS1 __attribute__((address_space(1)))
#define AS3 __attribute__((address_space(3)))

#if defined(__has_builtin)
#if __has_builtin(__builtin_amdgcn_global_load_async_to_lds_b128)
#define USE_ASYNC_LDS 1
#endif
#endif

// D = A(16x4) * B(4x16) + C(16x16), full fp32 (CDNA5 V_WMMA_F32_16X16X4_F32)
static __device__ inline v8f wmma_f32_16x16x4(v2f a, v2f b, v8f c) {
  return __builtin_amdgcn_wmma_f32_16x16x4_f32(
      /*neg_a=*/false, a, /*neg_b=*/false, b,
      /*c_mod=*/(short)0, c, /*reuse_a=*/false, /*reuse_b=*/false);
}

static __device__ inline float fast_rcp(float x) { return __builtin_amdgcn_rcpf(x); }
static __device__ inline float sigf(float x)  { return fast_rcp(1.0f + __expf(-x)); }
static __device__ inline float tanh_f(float x){ return 1.0f - 2.0f * fast_rcp(__expf(2.0f * x) + 1.0f); }

static __device__ inline float rl(float v, int lane) {
  return __builtin_bit_cast(float, __builtin_amdgcn_readlane(__builtin_bit_cast(int, v), lane));
}

// ---------------------------------------------------------------------------
// Scratch layout (floats)
// ---------------------------------------------------------------------------
// Mt    [256][32][2] : (W_up@W_N)^T pair-packed: element (k,c) at [k/2][c][k&1]
// cvec  [32]         : W_up @ b_N + b_up (padded)
// Wb0   [20][80]     : W_ih0 rearranged for WMMA-B layout, col n = unit*4+gate
// bias0 [80]         : b_ih0 + b_hh0, same n ordering
// seq   [512][257][20] : LSTM layer-0 inputs (fwd b, rev 256+b)
// xW0   [512][257][20][4] : precomputed layer-0 gate inputs (+bias)
// last  [512][20]    : final hidden of layer 1
#define OFF_MT    0
#define OFF_CV    16384
#define OFF_WB0   16416
#define OFF_BIAS0 18016
#define OFF_SEQ   18432
#define OFF_XW0   (18432 + 2631680)         // 2650112
#define OFF_LAST  (2650112 + 10526720)      // 13176832

// ---------------------------------------------------------------------------
// Prep 1: Mt (pair-packed) and cvec. grid 65 x 256
// ---------------------------------------------------------------------------
__global__ void k_prep1(const float* __restrict__ W_N, const float* __restrict__ b_N,
                        const float* __restrict__ W_up, const float* __restrict__ b_up,
                        float* __restrict__ Mt, float* __restrict__ cv) {
  if (blockIdx.x == 64) {
    int tid = threadIdx.x;
    if (tid < 32) {
      float acc = 0.0f;
      if (tid < L_) {
        for (int t = 0; t < S_; ++t) acc += W_up[tid * S_ + t] * b_N[t];
        acc += b_up[tid];
      }
      cv[tid] = acc;
    }
    return;
  }
  int idx = blockIdx.x * 256 + threadIdx.x;   // < 16384
  int s = idx >> 5, l = idx & 31;             // s = K row, l = output row (col)
  float acc = 0.0f;
  if (l < L_) {
    for (int t = 0; t < S_; ++t) acc += W_up[l * S_ + t] * W_N[t * S_ + s];
  }
  // pair-packed: (s, l) -> [s/2][l][s&1]
  Mt[(s >> 1) * 64 + l * 2 + (s & 1)] = acc;
}

// ---------------------------------------------------------------------------
// Prep 2: Wb0 (WMMA B-layout of W_ih0) + bias0. grid 1 x 256
// ---------------------------------------------------------------------------
__global__ void k_prep2(const float* __restrict__ W_ih0, const float* __restrict__ b_ih0,
                        const float* __restrict__ b_hh0,
                        float* __restrict__ Wb0, float* __restrict__ bias0) {
  for (int idx = threadIdx.x; idx < 1680; idx += 256) {
    if (idx < 1600) {
      int k = idx / G_, n = idx % G_;
      int unit = n >> 2, g = n & 3;
      Wb0[idx] = W_ih0[(g * L_ + unit) * L_ + k];
    } else {
      int n = idx - 1600;
      int unit = n >> 2, g = n & 3;
      bias0[n] = b_ih0[g * L_ + unit] + b_hh0[g * L_ + unit];
    }
  }
}

// ---------------------------------------------------------------------------
// Mu init: seq[s][0][l] = mu[l]. grid 40 x 256
// ---------------------------------------------------------------------------
__global__ void k_mu(const float* __restrict__ mu, float* __restrict__ seq) {
  int idx = blockIdx.x * 256 + threadIdx.x;
  int s = idx / L_, l = idx % L_;
  seq[(size_t)s * (T_ * L_) + l] = mu[l];
}

// ---------------------------------------------------------------------------
// Main GEMM (memory-bound, fp32 WMMA): x_up[b,l,f] = sum_s M[l,s]*x[b,s,f]+c[l]
// Mt staged into LDS via async-load-to-LDS (ASYNCcnt path) when available.
// grid 256 (one per batch) x 512 threads (16 waves; wave w -> f-tile w)
// ---------------------------------------------------------------------------
__global__ void k_gemm(const float* __restrict__ x, const float* __restrict__ Mt,
                       const float* __restrict__ cv, float* __restrict__ seq) {
  __shared__ __attribute__((aligned(16))) float MtL[S_ * 32];  // pair-packed, 64 KB
  __shared__ float cvL[32];
  int tid = threadIdx.x;

#ifdef USE_ASYNC_LDS
  for (int i = tid * 4; i < S_ * 32; i += 512 * 4) {
    __builtin_amdgcn_global_load_async_to_lds_b128(
        (AS1 v4i*)(Mt + i), (AS3 v4i*)(MtL + i), 0, 0);
  }
#if __has_builtin(__builtin_amdgcn_s_wait_asynccnt)
  __builtin_amdgcn_s_wait_asynccnt(0);
#else
  asm volatile("s_wait_asynccnt 0x0" ::: "memory");
#endif
#else
  {
    const float4* src4 = (const float4*)Mt;
    float4* dst4 = (float4*)MtL;
    for (int i = tid; i < (S_ * 32) / 4; i += 512) dst4[i] = src4[i];
  }
#endif
  if (tid < 32) cvL[tid] = cv[tid];
  __syncthreads();

  int b    = blockIdx.x;
  int wave = tid >> 5;
  int lane = tid & 31;
  int hi = lane >> 4, lo = lane & 15;
  int f0 = wave * 16;
  const float* xb = x + (size_t)b * (S_ * F_);
  const v2f* MtP = (const v2f*)MtL;   // [k/2][32] pairs

  v8f c0 = {0.f,0.f,0.f,0.f,0.f,0.f,0.f,0.f};
  v8f c1 = c0;
  for (int k0 = 0; k0 < S_; k0 += 4) {
    int r  = k0 + 2 * hi;            // even K row for this lane half
    int rp = (k0 >> 1) + hi;         // pair-row index
    v2f bf;
    bf.x = xb[(size_t)r * F_ + f0 + lo];
    bf.y = xb[(size_t)(r + 1) * F_ + f0 + lo];
    v2f a0 = MtP[rp * 32 + lo];          // one ds_load_b64
    v2f a1 = MtP[rp * 32 + 16 + lo];
    c0 = wmma_f32_16x16x4(a0, bf, c0);
    c1 = wmma_f32_16x16x4(a1, bf, c1);
  }

  int f = f0 + lo;
  int rb = 8 * hi;   // base output row for this lane half
  float* fwd = seq + (size_t)b * (T_ * L_) + (size_t)(f + 1) * L_;
  float* rev = seq + (size_t)(B_ + b) * (T_ * L_) + (size_t)(F_ - f) * L_;

  float4 o0 = {c0[0] + cvL[rb + 0], c0[1] + cvL[rb + 1],
               c0[2] + cvL[rb + 2], c0[3] + cvL[rb + 3]};
  float4 o1 = {c0[4] + cvL[rb + 4], c0[5] + cvL[rb + 5],
               c0[6] + cvL[rb + 6], c0[7] + cvL[rb + 7]};
  *(float4*)(fwd + rb)     = o0;
  *(float4*)(fwd + rb + 4) = o1;
  *(float4*)(rev + rb)     = o0;
  *(float4*)(rev + rb + 4) = o1;
  if (hi == 0) {   // rows 16..19 come from c1[0..3] on the low lane half
    float4 o2 = {c1[0] + cvL[16], c1[1] + cvL[17],
                 c1[2] + cvL[18], c1[3] + cvL[19]};
    *(float4*)(fwd + 16) = o2;
    *(float4*)(rev + 16) = o2;
  }
}

// ---------------------------------------------------------------------------
// Gate-input precompute (fp32 WMMA): xW0[m][n] = sum_l seq[m][l]*Wb0[l][n]+bias0[n]
// m = s*257+t (8224 M-tiles), n = unit*4+gate (5 N-tiles). grid 8224 x 160
// ---------------------------------------------------------------------------
__global__ void k_xw0(const float* __restrict__ seq, const float* __restrict__ Wb0,
                      const float* __restrict__ bias0, float* __restrict__ xW0) {
  __shared__ v2f  WbP[10 * G_];     // pair-packed rows (k,k+1)
  __shared__ float biasL[G_];
  int tid = threadIdx.x;
  for (int i = tid; i < 10 * G_; i += 160) {
    int kp = i / G_, n = i % G_;
    v2f w;
    w.x = Wb0[(2 * kp) * G_ + n];
    w.y = Wb0[(2 * kp + 1) * G_ + n];
    WbP[i] = w;
  }
  if (tid < G_) biasL[tid] = bias0[tid];
  __syncthreads();

  int wave = tid >> 5;
  int lane = tid & 31;
  int hi = lane >> 4, lo = lane & 15;
  int n0 = wave * 16;
  int m0 = blockIdx.x * 16;

  v8f c = {0.f,0.f,0.f,0.f,0.f,0.f,0.f,0.f};
#pragma unroll
  for (int k0 = 0; k0 < L_; k0 += 4) {
    int k  = k0 + 2 * hi;
    int kp = (k0 >> 1) + hi;
    v2f a = *(const v2f*)(seq + (size_t)(m0 + lo) * L_ + k);  // one b64 load
    v2f b = WbP[kp * G_ + n0 + lo];                           // one ds_load_b64
    c = wmma_f32_16x16x4(a, b, c);
  }
  int n = n0 + lo;
  float bias = biasL[n];
#pragma unroll
  for (int v = 0; v < 8; ++v) {
    int m = m0 + v + 8 * hi;
    xW0[(size_t)m * G_ + n] = c[v] + bias;
  }
}

// ---------------------------------------------------------------------------
// Fused 2-layer LSTM scan: one wave32 per sequence, lane = hidden unit.
// Recurrent weights in per-lane VGPRs; h broadcast via v_readlane (no barriers).
// grid 64 x 256 (8 waves/block -> 8 sequences/block)
// ---------------------------------------------------------------------------
__global__ void __launch_bounds__(256, 1)
k_scan(const float* __restrict__ xW0,
       const float* __restrict__ W_hh0,
       const float* __restrict__ W_ih1, const float* __restrict__ W_hh1,
       const float* __restrict__ b_ih1, const float* __restrict__ b_hh1,
       float* __restrict__ last) {
  int s    = blockIdx.x * 8 + (threadIdx.x >> 5);
  int lane = threadIdx.x & 31;
  int lc   = lane < L_ ? lane : (L_ - 1);

  float wh0[4][L_], wi1[4][L_], wh1[4][L_], bz1[4];
#pragma unroll
  for (int g = 0; g < 4; ++g) {
    bz1[g] = b_ih1[g * L_ + lc] + b_hh1[g * L_ + lc];
#pragma unroll
    for (int k = 0; k < L_; ++k) {
      wh0[g][k] = W_hh0[(g * L_ + lc) * L_ + k];
      wi1[g][k] = W_ih1[(g * L_ + lc) * L_ + k];
      wh1[g][k] = W_hh1[(g * L_ + lc) * L_ + k];
    }
  }

  float h0 = 0.f, c0 = 0.f, h1 = 0.f, c1 = 0.f;
  const float* xwb = xW0 + (size_t)s * (T_ * G_) + lc * 4;

  for (int t = 0; t < T_; ++t) {
    const float4 xw = *reinterpret_cast<const float4*>(xwb + (size_t)t * G_);
    if (t + 1 < T_) __builtin_prefetch(xwb + (size_t)(t + 1) * G_, 0, 0);

    // ---- layer 0: gates = xW0 + h0 @ W_hh0^T
    float a0 = xw.x, a1 = xw.y, a2 = xw.z, a3 = xw.w;
#pragma unroll
    for (int k = 0; k < L_; ++k) {
      float hk = rl(h0, k);
      a0 += hk * wh0[0][k];
      a1 += hk * wh0[1][k];
      a2 += hk * wh0[2][k];
      a3 += hk * wh0[3][k];
    }
    float ig = sigf(a0), fg = sigf(a1), gg = tanh_f(a2), og = sigf(a3);
    c0 = fg * c0 + ig * gg;
    h0 = og * tanh_f(c0);

    // ---- layer 1: gates = bias1 + h0 @ W_ih1^T + h1 @ W_hh1^T
    a0 = bz1[0]; a1 = bz1[1]; a2 = bz1[2]; a3 = bz1[3];
#pragma unroll
    for (int k = 0; k < L_; ++k) {
      float xk = rl(h0, k);
      float hk = rl(h1, k);
      a0 += xk * wi1[0][k] + hk * wh1[0][k];
      a1 += xk * wi1[1][k] + hk * wh1[1][k];
      a2 += xk * wi1[2][k] + hk * wh1[2][k];
      a3 += xk * wi1[3][k] + hk * wh1[3][k];
    }
    ig = sigf(a0); fg = sigf(a1); gg = tanh_f(a2); og = sigf(a3);
    c1 = fg * c1 + ig * gg;
    h1 = og * tanh_f(c1);
  }

  if (lane < L_) last[(size_t)s * L_ + lane] = h1;
}

// ---------------------------------------------------------------------------
// Head: y[b] = W_view . concat(last_fwd[b], last_rev[b]) + b_view. grid 1 x 256
// ---------------------------------------------------------------------------
__global__ void k_head(const float* __restrict__ last, const float* __restrict__ W_view,
                       const float* __restrict__ b_view, float* __restrict__ out) {
  int b = threadIdx.x;
  float acc = b_view[0];
#pragma unroll
  for (int l = 0; l < L_; ++l) {
    acc += W_view[l]      * last[(size_t)b * L_ + l];
    acc += W_view[L_ + l] * last[(size_t)(B_ + b) * L_ + l];
  }
  out[b] = acc;
}

// ---------------------------------------------------------------------------
extern "C" void kernel_launch(void* const* d_in, const int* in_sizes, int n_in,
                              void* d_out, int out_size, void* d_ws, size_t ws_size,
                              hipStream_t stream) {
  const float* x      = (const float*)d_in[0];
  const float* W_N    = (const float*)d_in[1];
  const float* b_N    = (const float*)d_in[2];
  const float* W_up   = (const float*)d_in[3];
  const float* b_up   = (const float*)d_in[4];
  const float* mu     = (const float*)d_in[5];
  const float* W_ih0  = (const float*)d_in[6];
  const float* W_hh0  = (const float*)d_in[7];
  const float* b_ih0  = (const float*)d_in[8];
  const float* b_hh0  = (const float*)d_in[9];
  const float* W_ih1  = (const float*)d_in[10];
  const float* W_hh1  = (const float*)d_in[11];
  const float* b_ih1  = (const float*)d_in[12];
  const float* b_hh1  = (const float*)d_in[13];
  const float* W_view = (const float*)d_in[14];
  const float* b_view = (const float*)d_in[15];
  float* out = (float*)d_out;

  float* wsf   = (float*)d_ws;
  float* Mt    = wsf + OFF_MT;
  float* cv    = wsf + OFF_CV;
  float* Wb0   = wsf + OFF_WB0;
  float* bias0 = wsf + OFF_BIAS0;
  float* seq   = wsf + OFF_SEQ;
  float* xW0   = wsf + OFF_XW0;
  float* last  = wsf + OFF_LAST;

  k_prep1<<<65, 256, 0, stream>>>(W_N, b_N, W_up, b_up, Mt, cv);
  k_prep2<<<1, 256, 0, stream>>>(W_ih0, b_ih0, b_hh0, Wb0, bias0);
  k_mu<<<40, 256, 0, stream>>>(mu, seq);
  k_gemm<<<B_, 512, 0, stream>>>(x, Mt, cv, seq);
  k_xw0<<<(NSEQ * T_) / 16, 160, 0, stream>>>(seq, Wb0, bias0, xW0);
  k_scan<<<NSEQ / 8, 256, 0, stream>>>(xW0, W_hh0, W_ih1, W_hh1, b_ih1, b_hh1, last);
  k_head<<<1, 256, 0, stream>>>(last, W_view, b_view, out);
}